// LSTM_51719996178893
// MI455X (gfx1250) — compile-verified
//
#include <hip/hip_runtime.h>
#include <hip/hip_bf16.h>
#include <math.h>

typedef float v2f __attribute__((ext_vector_type(2)));
typedef float v8f __attribute__((ext_vector_type(8)));

#define LSTM_N 8192
#define LSTM_T 256
#define LSTM_D 44
#define LSTM_H 8
#define LSTM_G 1024
#define KSLICES 11   // D = 44 = 11 * 4

__device__ __forceinline__ float sigm(float v) {
    // 1/(1+e^-v) with fast v_rcp_f32 (avoids IEEE div expansion)
    return __builtin_amdgcn_rcpf(1.0f + __expf(-v));
}

__device__ __forceinline__ float tanh_f(float v) {
#if __has_builtin(__builtin_amdgcn_tanhf)
    return __builtin_amdgcn_tanhf(v);   // native V_TANH_F32 on gfx1250
#else
    const float e = __expf(2.0f * v);
    return (e - 1.0f) * __builtin_amdgcn_rcpf(e + 1.0f);
#endif
}

// ---------------------------------------------------------------------------
// Kernel 1: zero the segment-sum accumulator (G*H floats in workspace)
// ---------------------------------------------------------------------------
__global__ void zero_seg_kernel(float* __restrict__ seg, int n) {
    int i = blockIdx.x * blockDim.x + threadIdx.x;
    if (i < n) seg[i] = 0.0f;
}

// ---------------------------------------------------------------------------
// Kernel 2: fused WMMA LSTM over 16 paths per wave (one wave per block).
// Lane L and lane L+16 both own path (blockIdx.x*16 + L%16): the two halves
// supply the K / K+2 components of the f32 WMMA A-operand. Each lane computes
// the hidden units in a half-dependent permuted register order
//   u[j] = {0,1,4,5,2,3,6,7}[j] ^ (2*half)
// so the recurrent WMMA A-operand is always (h[0],h[1]) and (h[2],h[3]) —
// fixed registers, no per-step lane-varying selects. The permutation is
// folded into LDS gate-read addresses and seg atomic addresses (hoisted).
// Gate tiles: acc0 -> gates 0..15 (i|f), acc1 -> gates 16..31 (g|o).
// ---------------------------------------------------------------------------
__global__ __launch_bounds__(32) void lstm_fused_kernel(
    const float* __restrict__ x,        // [N, T, D]
    const float* __restrict__ W_ih,     // [4H, D]
    const float* __restrict__ W_hh,     // [4H, H]
    const float* __restrict__ b_ih,     // [4H]
    const float* __restrict__ b_hh,     // [4H]
    const int*   __restrict__ last_idx, // [N]
    const int*   __restrict__ group_ids,// [N]
    float* __restrict__ seg)            // [G, H] atomic accumulator
{
    __shared__ float gl[16 * 33];       // 16x32 gate tile, padded stride

    const int lane = threadIdx.x;       // 0..31
    const int row  = lane & 15;         // path row within tile / B column
    const int half = lane >> 4;         // 0: K even pair, 1: K+2 pair
    const int path = blockIdx.x * 16 + row;

    // ---- B operands (weights, transposed into WMMA B layout), in registers.
    // Slice k, VGPR0 holds K = 4k + 2*half, VGPR1 holds K = 4k + 2*half + 1.
    v2f Bi0[KSLICES], Bi1[KSLICES];
#pragma unroll
    for (int k = 0; k < KSLICES; ++k) {
        const int d = 4 * k + 2 * half;
        Bi0[k] = v2f{W_ih[row * LSTM_D + d],        W_ih[row * LSTM_D + d + 1]};
        Bi1[k] = v2f{W_ih[(16 + row) * LSTM_D + d], W_ih[(16 + row) * LSTM_D + d + 1]};
    }
    v2f Bh0[2], Bh1[2];
#pragma unroll
    for (int s = 0; s < 2; ++s) {
        const int d = 4 * s + 2 * half;
        Bh0[s] = v2f{W_hh[row * LSTM_H + d],        W_hh[row * LSTM_H + d + 1]};
        Bh1[s] = v2f{W_hh[(16 + row) * LSTM_H + d], W_hh[(16 + row) * LSTM_H + d + 1]};
    }
    const float bias0 = b_ih[row]      + b_hh[row];        // column = gate N
    const float bias1 = b_ih[16 + row] + b_hh[16 + row];

    const float* xrow = x + (size_t)path * LSTM_T * LSTM_D;
    const int lidx = last_idx[path];
    const int gid  = group_ids[path];

    // Per-lane permuted logical unit ids and hoisted addresses.
    const int twoH = 2 * half;
    const float* gp[LSTM_H];   // LDS gate base pointer per register slot
    float*       sp[LSTM_H];   // seg atomic target per register slot
    {
        const int ubase[LSTM_H] = {0, 1, 4, 5, 2, 3, 6, 7};
#pragma unroll
        for (int j = 0; j < LSTM_H; ++j) {
            const int u = ubase[j] ^ twoH;          // logical hidden unit
            gp[j] = &gl[row * 33 + u];
            sp[j] = &seg[gid * LSTM_H + u];
        }
    }

    float h[LSTM_H], c[LSTM_H];
#pragma unroll
    for (int j = 0; j < LSTM_H; ++j) { h[j] = 0.0f; c[j] = 0.0f; }

    for (int t = 0; t < LSTM_T; ++t) {
        const float* xp = xrow + t * LSTM_D + twoH;
        if (t + 1 < LSTM_T)
            __builtin_prefetch(xrow + (t + 1) * LSTM_D, 0, 1);

        v8f acc0 = {bias0, bias0, bias0, bias0, bias0, bias0, bias0, bias0};
        v8f acc1 = {bias1, bias1, bias1, bias1, bias1, bias1, bias1, bias1};

        // Input contribution: X_t[16x44] @ W_ih^T, 11 K-slices of f32 WMMA.
#pragma unroll
        for (int k = 0; k < KSLICES; ++k) {
            v2f a = *(const v2f*)(xp + 4 * k);
            acc0 = __builtin_amdgcn_wmma_f32_16x16x4_f32(
                false, a, false, Bi0[k], (short)0, acc0, false, false);
            acc1 = __builtin_amdgcn_wmma_f32_16x16x4_f32(
                false, a, false, Bi1[k], (short)0, acc1, false, false);
        }
        // Recurrent contribution: H[16x8] @ W_hh^T, 2 K-slices.
        // Register slots 0..3 hold exactly the logical units this lane-half
        // must supply (u = 2*half,2*half+1,4+2*half,5+2*half).
        {
            v2f a0 = v2f{h[0], h[1]};
            v2f a1 = v2f{h[2], h[3]};
            acc0 = __builtin_amdgcn_wmma_f32_16x16x4_f32(
                false, a0, false, Bh0[0], (short)0, acc0, false, false);
            acc1 = __builtin_amdgcn_wmma_f32_16x16x4_f32(
                false, a0, false, Bh1[0], (short)0, acc1, false, false);
            acc0 = __builtin_amdgcn_wmma_f32_16x16x4_f32(
                false, a1, false, Bh0[1], (short)0, acc0, false, false);
            acc1 = __builtin_amdgcn_wmma_f32_16x16x4_f32(
                false, a1, false, Bh1[1], (short)0, acc1, false, false);
        }

        // Accumulator -> LDS (C/D layout: VGPR r, lane L => M=r+8*half, N=L%16).
#pragma unroll
        for (int r = 0; r < 8; ++r) {
            const int m = r + 8 * half;
            gl[m * 33 + row]      = acc0[r];
            gl[m * 33 + 16 + row] = acc1[r];
        }
        __syncthreads();

        // Per-lane LSTM cell update (units in permuted register order).
#pragma unroll
        for (int j = 0; j < LSTM_H; ++j) {
            const float ig = gp[j][0];
            const float fg = gp[j][8];
            const float gg = gp[j][16];
            const float og = gp[j][24];
            c[j] = sigm(fg) * c[j] + sigm(ig) * tanh_f(gg);
            h[j] = sigm(og) * tanh_f(c[j]);
        }
        __syncthreads();

        if (t == lidx && half == 0) {
#pragma unroll
            for (int j = 0; j < LSTM_H; ++j)
                atomicAdd(sp[j], h[j]);
        }
    }
}

// ---------------------------------------------------------------------------
// Kernel 3: logits = seg @ fc_W^T + fc_b, softmax over the 1024 groups.
// ---------------------------------------------------------------------------
__global__ __launch_bounds__(LSTM_G) void head_softmax_kernel(
    const float* __restrict__ seg,     // [G, H]
    const float* __restrict__ fc_W,    // [1, H]
    const float* __restrict__ fc_b,    // [1]
    float* __restrict__ out)           // [1, G]
{
    __shared__ float red[LSTM_G];
    const int g = threadIdx.x;

    float logit = fc_b[0];
#pragma unroll
    for (int j = 0; j < LSTM_H; ++j)
        logit += seg[g * LSTM_H + j] * fc_W[j];

    red[g] = logit;
    __syncthreads();
    for (int s = LSTM_G / 2; s > 0; s >>= 1) {
        if (g < s) red[g] = fmaxf(red[g], red[g + s]);
        __syncthreads();
    }
    const float m = red[0];
    __syncthreads();

    const float e = __expf(logit - m);
    red[g] = e;
    __syncthreads();
    for (int s = LSTM_G / 2; s > 0; s >>= 1) {
        if (g < s) red[g] += red[g + s];
        __syncthreads();
    }
    out[g] = e / red[0];
}

// ---------------------------------------------------------------------------
extern "C" void kernel_launch(void* const* d_in, const int* in_sizes, int n_in,
                              void* d_out, int out_size, void* d_ws, size_t ws_size,
                              hipStream_t stream) {
    const float* x         = (const float*)d_in[0];
    const float* W_ih      = (const float*)d_in[1];
    const float* W_hh      = (const float*)d_in[2];
    const float* b_ih      = (const float*)d_in[3];
    const float* b_hh      = (const float*)d_in[4];
    const float* fc_W      = (const float*)d_in[5];
    const float* fc_b      = (const float*)d_in[6];
    const int*   last_idx  = (const int*)d_in[7];
    const int*   group_ids = (const int*)d_in[8];

    float* seg = (float*)d_ws;                 // G*H = 8192 floats (32 KB)
    float* out = (float*)d_out;                // [1, G] float32

    zero_seg_kernel<<<(LSTM_G * LSTM_H + 255) / 256, 256, 0, stream>>>(
        seg, LSTM_G * LSTM_H);

    lstm_fused_kernel<<<LSTM_N / 16, 32, 0, stream>>>(
        x, W_ih, W_hh, b_ih, b_hh, last_idx, group_ids, seg);

    head_softmax_kernel<<<1, LSTM_G, 0, stream>>>(seg, fc_W, fc_b, out);
}